// SPLLoss_11922829213864
// MI455X (gfx1250) — compile-verified
//
#include <hip/hip_runtime.h>
#include <math.h>

#ifndef __has_builtin
#define __has_builtin(x) 0
#endif

#if __has_builtin(__builtin_amdgcn_global_load_async_to_lds_b128)
#define HAVE_ASYNC_LDS 1
#else
#define HAVE_ASYNC_LDS 0
#endif

#define TPB 256
#define NWAVES (TPB / 32)

// 16-byte vector type matching the async-LDS builtin's parameter type:
//   '__attribute__((__vector_size__(4 * sizeof(int)))) int __device__ *'
typedef int v4i __attribute__((vector_size(4 * sizeof(int))));
typedef __attribute__((address_space(1))) v4i* g_v4i_p;   // global (AS1)
typedef __attribute__((address_space(3))) v4i* l_v4i_p;   // LDS (AS3)

__device__ __forceinline__ float wave_reduce_max(float v) {
#pragma unroll
    for (int off = 16; off > 0; off >>= 1)
        v = fmaxf(v, __shfl_xor(v, off, 32));
    return v;
}

__device__ __forceinline__ float wave_reduce_sum(float v) {
#pragma unroll
    for (int off = 16; off > 0; off >>= 1)
        v += __shfl_xor(v, off, 32);
    return v;
}

// One workgroup per row. Stages the whole 128KB row into LDS via async
// global->LDS b128 copies, then does a stable two-pass softmax from LDS
// (max pass + sum-exp pass), gathers the two target log-probs from LDS,
// and writes the per-row SPL-weighted mixup loss + mask to workspace.
__global__ void __launch_bounds__(TPB)
spl_row_kernel(const float* __restrict__ input,
               const int* __restrict__ targeta,
               const int* __restrict__ targetb,
               const float* __restrict__ thr_p,
               const float* __restrict__ lam_p,
               float* __restrict__ ws_loss,
               float* __restrict__ ws_mask,
               int C)
{
    extern __shared__ float smem[];          // C floats (row staged here)
    __shared__ float red[NWAVES];

    const int row  = blockIdx.x;
    const int tid  = threadIdx.x;
    const int lane = tid & 31;
    const int wid  = tid >> 5;
    const float* rowp = input + (size_t)row * (size_t)C;

    // ---- stage row HBM -> LDS (16B per lane per issue) ----
    for (int idx = tid * 4; idx < C; idx += TPB * 4) {
#if HAVE_ASYNC_LDS
        __builtin_amdgcn_global_load_async_to_lds_b128(
            (g_v4i_p)(rowp + idx),
            (l_v4i_p)(smem + idx),
            0, 0);
#else
        *(float4*)(smem + idx) = *(const float4*)(rowp + idx);
#endif
    }
#if HAVE_ASYNC_LDS
#if __has_builtin(__builtin_amdgcn_s_wait_asynccnt)
    __builtin_amdgcn_s_wait_asynccnt(0);
#else
    asm volatile("s_wait_asynccnt 0" ::: "memory");
#endif
#endif
    __syncthreads();

    const float4* s4 = (const float4*)smem;
    const int C4 = C >> 2;

    // ---- pass 1: row max (from LDS) ----
    float m = -3.402823466e38f;
    for (int i = tid; i < C4; i += TPB) {
        float4 x = s4[i];
        m = fmaxf(m, fmaxf(fmaxf(x.x, x.y), fmaxf(x.z, x.w)));
    }
    m = wave_reduce_max(m);
    if (lane == 0) red[wid] = m;
    __syncthreads();
    float t = (lane < NWAVES) ? red[lane] : -3.402823466e38f;
    const float M = wave_reduce_max(t);      // every thread has row max
    __syncthreads();                          // protect red[] reuse

    // ---- pass 2: sum exp(x - M) (from LDS) ----
    float s = 0.0f;
    for (int i = tid; i < C4; i += TPB) {
        float4 x = s4[i];
        s += __expf(x.x - M) + __expf(x.y - M) +
             __expf(x.z - M) + __expf(x.w - M);
    }
    s = wave_reduce_sum(s);
    if (lane == 0) red[wid] = s;
    __syncthreads();
    float u = (lane < NWAVES) ? red[lane] : 0.0f;
    const float S = wave_reduce_sum(u);

    // ---- per-row scalar epilogue ----
    if (tid == 0) {
        const int ia = targeta[row];
        const int ib = targetb[row];
        const float lse  = M + __logf(S);            // logsumexp
        const float ce_a = lse - smem[ia];           // -logp[ia]
        const float ce_b = lse - smem[ib];           // -logp[ib]
        const float lam  = lam_p[0];
        const float thr  = thr_p[0];
        const float super_loss = lam * ce_a + (1.0f - lam) * ce_b;
        const float v = (1.0f + __expf(-thr)) / (1.0f + __expf(super_loss - thr));
        const float msk = (ia != ib) ? 1.0f : 0.0f;
        ws_loss[row] = super_loss * v * msk;
        ws_mask[row] = msk;
    }
}

// Deterministic fixed-order final reduction: out = sum(loss*v*mask) / sum(mask)
__global__ void __launch_bounds__(TPB)
spl_finalize_kernel(const float* __restrict__ ws_loss,
                    const float* __restrict__ ws_mask,
                    float* __restrict__ out, int N)
{
    __shared__ float ra[NWAVES], rb[NWAVES];
    const int tid  = threadIdx.x;
    const int lane = tid & 31;
    const int wid  = tid >> 5;

    float a = 0.0f, b = 0.0f;
    for (int i = tid; i < N; i += TPB) {
        a += ws_loss[i];
        b += ws_mask[i];
    }
    a = wave_reduce_sum(a);
    b = wave_reduce_sum(b);
    if (lane == 0) { ra[wid] = a; rb[wid] = b; }
    __syncthreads();
    if (wid == 0) {
        float x = (lane < NWAVES) ? ra[lane] : 0.0f;
        float y = (lane < NWAVES) ? rb[lane] : 0.0f;
        x = wave_reduce_sum(x);
        y = wave_reduce_sum(y);
        if (lane == 0) out[0] = x / y;
    }
}

extern "C" void kernel_launch(void* const* d_in, const int* in_sizes, int n_in,
                              void* d_out, int out_size, void* d_ws, size_t ws_size,
                              hipStream_t stream) {
    const float* input   = (const float*)d_in[0];   // [N, C] f32
    const int*   targeta = (const int*)d_in[1];     // [N]
    const int*   targetb = (const int*)d_in[2];     // [N]
    const float* thr     = (const float*)d_in[3];   // scalar
    const float* lam     = (const float*)d_in[4];   // scalar

    const int N = in_sizes[1];
    const int C = in_sizes[0] / N;

    float* ws_loss = (float*)d_ws;
    float* ws_mask = ws_loss + N;

    const size_t lds_bytes = (size_t)C * sizeof(float);  // 128000 B < 320 KB WGP LDS
    spl_row_kernel<<<N, TPB, lds_bytes, stream>>>(input, targeta, targetb,
                                                  thr, lam, ws_loss, ws_mask, C);
    spl_finalize_kernel<<<1, TPB, 0, stream>>>(ws_loss, ws_mask, (float*)d_out, N);
}